// SimpleAttentionLayer_11003706212938
// MI455X (gfx1250) — compile-verified
//
#include <hip/hip_runtime.h>
#include <math.h>

typedef __attribute__((ext_vector_type(2))) float v2f;
typedef __attribute__((ext_vector_type(8))) float v8f;

#define WMMA_F32(a, b, c) \
  __builtin_amdgcn_wmma_f32_16x16x4_f32(false, (a), false, (b), (short)0, (c), false, false)

static constexpr int Bsz    = 8;
static constexpr int Bpad   = 16;         // batch padded to WMMA tile
static constexpr int Hdim   = 2048;
static constexpr int NH     = 16;
static constexpr int HD     = 128;
static constexpr int Pctx   = 8192;
static constexpr int QKV_N  = 3 * Hdim;   // 6144
static constexpr int KSPLIT = 4;          // K-split for the two GEMMs
static constexpr int NSPLIT = 64;         // attention splits per head
static constexpr int CHUNK  = Pctx / NSPLIT; // 128 positions per wave-split
static constexpr float SCALE = 0.08838834764831845f; // 1/sqrt(128)

// ---------------------------------------------------------------------------
// Copy x (8 rows) into a 16-row zero-padded buffer.
// ---------------------------------------------------------------------------
__global__ void attn_copy_pad16(const float* __restrict__ src,
                                float* __restrict__ dst, int K) {
  int idx = blockIdx.x * blockDim.x + threadIdx.x;
  if (idx >= Bpad * K) return;
  int row = idx / K;
  dst[idx] = (row < Bsz) ? src[idx] : 0.f;
}

// Zero rows 8..15 of a 16-row buffer (producers fill rows 0..7 later/earlier).
__global__ void attn_zero_tail(float* __restrict__ dst, int K) {
  int idx = blockIdx.x * blockDim.x + threadIdx.x;
  if (idx >= Bsz * K) return;
  dst[(size_t)Bsz * K + idx] = 0.f;
}

// ---------------------------------------------------------------------------
// y^T = W (N x 2048) * x^T (2048 x 16, rows 8..15 of x are zero-padded).
// One wave computes a 16(feature) x 16(batch) tile over K/KSPLIT.
// Inner loop: 2 unconditional global_load_b64 + 1 v_wmma per K-step.
// ---------------------------------------------------------------------------
__global__ void attn_gemm_xwT_partial(const float* __restrict__ xpad,
                                      const float* __restrict__ w,
                                      float* __restrict__ part, int N) {
  const int K    = Hdim;
  const int wave = threadIdx.x >> 5;
  const int lane = threadIdx.x & 31;
  const int row  = lane & 15;
  const int half = lane >> 4;
  const int n0   = (blockIdx.x * 8 + wave) * 16;
  const int split = blockIdx.y;
  const int ks0  = split * (K / KSPLIT);

  v8f c = {};
  const float* wp = w + (size_t)(n0 + row) * K + ks0 + 2 * half;
  const float* xp = xpad + (size_t)row * K + ks0 + 2 * half;

#pragma unroll 8
  for (int it = 0; it < K / KSPLIT / 4; ++it) {
    v2f a = *(const v2f*)(wp + 4 * it);   // A: W[M=feat][K]
    v2f b = *(const v2f*)(xp + 4 * it);   // B: x^T[K][N=batch]
    c = WMMA_F32(a, b, c);
  }
  if (row < Bsz) {
#pragma unroll
    for (int d = 0; d < 8; ++d)
      part[((size_t)split * Bsz + row) * N + n0 + d + 8 * half] = c[d];
  }
}

__global__ void attn_reduce_bias(const float* __restrict__ part,
                                 const float* __restrict__ bias,
                                 float* __restrict__ out, int N) {
  int idx = blockIdx.x * blockDim.x + threadIdx.x;
  if (idx >= Bsz * N) return;
  int b = idx / N, n = idx % N;
  float s = bias[n];
#pragma unroll
  for (int sp = 0; sp < KSPLIT; ++sp) s += part[((size_t)sp * Bsz + b) * N + n];
  out[idx] = s;
}

// ---------------------------------------------------------------------------
// Flash-decoding split: one wave per (head, split of 128 cached positions).
// Online softmax over 8 pos-tiles of 16; both GEMMs use f32 WMMA 16x16x4.
// qkv buffer is 16-row zero-padded so Q-fragment loads are unconditional.
// ---------------------------------------------------------------------------
__global__ void attn_split_kernel(const float* __restrict__ qkv,
                                  const float* __restrict__ ck,
                                  const float* __restrict__ cv,
                                  float* __restrict__ wm,
                                  float* __restrict__ wl,
                                  float* __restrict__ wacc) {
  __shared__ float pmat[8 * 256];           // 16x16 P tile per wave (wave-private)
  const int wave = threadIdx.x >> 5;
  const int lane = threadIdx.x & 31;
  const int row  = lane & 15;
  const int half = lane >> 4;
  const int h    = blockIdx.x;
  const int s    = blockIdx.y * 8 + wave;
  const int pbase = s * CHUNK;
  float* lp = pmat + wave * 256;

  // Preload Q fragments: B-layout, K(dim) = 4k + 2*half + j, N(batch) = row.
  v2f qf[32];
#pragma unroll
  for (int k = 0; k < 32; ++k)
    qf[k] = *(const v2f*)(qkv + (size_t)row * QKV_N + h * HD + 4 * k + 2 * half);

  float m = -INFINITY, l = 0.f;
  v8f acc[8];
#pragma unroll
  for (int t = 0; t < 8; ++t) acc[t] = {};

  for (int t = 0; t < CHUNK / 16; ++t) {
    const int p0 = pbase + t * 16;

    // --- scores tile: A = K rows (M=pos, K=dim), B = Q^T -> D[pos][batch]
    v8f sc = {};
    const float* kp = ck + (size_t)(p0 + row) * (NH * HD) + h * HD + 2 * half;
#pragma unroll
    for (int k = 0; k < 32; ++k) {
      v2f a = *(const v2f*)(kp + 4 * k);
      sc = WMMA_F32(a, qf[k], sc);
    }
#pragma unroll
    for (int d = 0; d < 8; ++d) sc[d] *= SCALE;

    // --- online softmax (per batch = row; lanes row and row+16 hold pos 0-7 / 8-15)
    float tm = sc[0];
#pragma unroll
    for (int d = 1; d < 8; ++d) tm = fmaxf(tm, sc[d]);
    tm = fmaxf(tm, __shfl_xor(tm, 16, 32));
    float mnew = fmaxf(m, tm);
    float fac  = __expf(m - mnew);
    float p[8];
    float ls = 0.f;
#pragma unroll
    for (int d = 0; d < 8; ++d) { p[d] = __expf(sc[d] - mnew); ls += p[d]; }
    ls += __shfl_xor(ls, 16, 32);
    l = l * fac + ls;
    m = mnew;

    // rescale acc: acc batch index is d + 8*half -> fetch that batch's factor
    float facv[8];
#pragma unroll
    for (int d = 0; d < 8; ++d) facv[d] = __shfl(fac, d + 8 * half, 32);
#pragma unroll
    for (int t2 = 0; t2 < 8; ++t2) {
#pragma unroll
      for (int d = 0; d < 8; ++d) acc[t2][d] *= facv[d];
    }

    // --- bounce P through LDS to re-layout D[pos][batch] -> A[batch][pos]
#pragma unroll
    for (int d = 0; d < 8; ++d) lp[(d + 8 * half) * 16 + row] = p[d];
    // wave-private region; LDS ops from one wave complete in order (DScnt)

    // --- acc += P^T x V : A(M=batch,K=pos), B = V(K=pos, N=dim)
#pragma unroll
    for (int k = 0; k < 4; ++k) {
      v2f a;
      a.x = lp[(4 * k + 2 * half + 0) * 16 + row];
      a.y = lp[(4 * k + 2 * half + 1) * 16 + row];
      const float* vp = cv + (size_t)(p0 + 4 * k + 2 * half) * (NH * HD) + h * HD + row;
#pragma unroll
      for (int t2 = 0; t2 < 8; ++t2) {
        v2f b;
        b.x = vp[t2 * 16];
        b.y = vp[(size_t)(NH * HD) + t2 * 16];
        acc[t2] = WMMA_F32(a, b, acc[t2]);
      }
    }
  }

  // --- write split partials
  const size_t base = ((size_t)h * NSPLIT + s) * Bsz;
  if (lane < Bsz) {
    wm[base + lane] = m;
    wl[base + lane] = l;
  }
  if (half == 0) {
#pragma unroll
    for (int d = 0; d < 8; ++d) {
#pragma unroll
      for (int t2 = 0; t2 < 8; ++t2)
        wacc[(base + d) * HD + t2 * 16 + row] = acc[t2][d];
    }
  }
}

// ---------------------------------------------------------------------------
// Merge NSPLIT partials per (h,b) + fold in the per-batch new-token key/value.
// Block = 128 threads (one per head-dim).
// ---------------------------------------------------------------------------
__global__ void attn_merge_kernel(const float* __restrict__ qkv,
                                  const float* __restrict__ wm,
                                  const float* __restrict__ wl,
                                  const float* __restrict__ wacc,
                                  float* __restrict__ aout) {
  __shared__ float red[128];
  const int h = blockIdx.x >> 3;
  const int b = blockIdx.x & 7;
  const int d = threadIdx.x;
  const float* qv = qkv + (size_t)b * QKV_N + h * HD;

  // s_new = q . k_new * scale  (new token's key differs per batch)
  red[d] = qv[d] * qv[Hdim + d];
  __syncthreads();
  for (int off = 64; off > 0; off >>= 1) {
    if (d < off) red[d] += red[d + off];
    __syncthreads();
  }
  const float snew = red[0] * SCALE;

  float M = snew;
  for (int s = 0; s < NSPLIT; ++s)
    M = fmaxf(M, wm[(size_t)h * NSPLIT * Bsz + (size_t)s * Bsz + b]);

  float wnew = __expf(snew - M);
  float num = wnew * qv[2 * Hdim + d];   // v_new contribution
  float den = wnew;
  for (int s = 0; s < NSPLIT; ++s) {
    const size_t i = (size_t)h * NSPLIT * Bsz + (size_t)s * Bsz + b;
    const float w = __expf(wm[i] - M);
    num += w * wacc[i * HD + d];
    den += w * wl[i];
  }
  aout[(size_t)b * Hdim + h * HD + d] = num / den;
}

// ---------------------------------------------------------------------------
extern "C" void kernel_launch(void* const* d_in, const int* in_sizes, int n_in,
                              void* d_out, int out_size, void* d_ws, size_t ws_size,
                              hipStream_t stream) {
  const float* x     = (const float*)d_in[0];
  const float* ck    = (const float*)d_in[1];
  const float* cv    = (const float*)d_in[2];
  const float* wqkv  = (const float*)d_in[3];
  const float* bqkv  = (const float*)d_in[4];
  const float* wproj = (const float*)d_in[5];
  const float* bproj = (const float*)d_in[6];
  (void)in_sizes; (void)n_in; (void)out_size; (void)ws_size;

  float* ws        = (float*)d_ws;
  float* x_pad     = ws;                                      // 16*2048
  float* qkv_part  = x_pad + (size_t)Bpad * Hdim;             // 4*8*6144
  float* qkv_pad   = qkv_part + (size_t)KSPLIT * Bsz * QKV_N; // 16*6144
  float* wm        = qkv_pad + (size_t)Bpad * QKV_N;          // 16*64*8
  float* wl        = wm + (size_t)NH * NSPLIT * Bsz;          // 16*64*8
  float* wacc      = wl + (size_t)NH * NSPLIT * Bsz;          // 16*64*8*128
  float* aout_pad  = wacc + (size_t)NH * NSPLIT * Bsz * HD;   // 16*2048
  float* proj_part = aout_pad + (size_t)Bpad * Hdim;          // 4*8*2048

  // 0) Build zero-padded 16-row operands (cheap; keeps GEMM loops branch-free)
  attn_copy_pad16<<<(Bpad * Hdim + 255) / 256, 256, 0, stream>>>(x, x_pad, Hdim);
  attn_zero_tail<<<(Bsz * QKV_N + 255) / 256, 256, 0, stream>>>(qkv_pad, QKV_N);
  attn_zero_tail<<<(Bsz * Hdim + 255) / 256, 256, 0, stream>>>(aout_pad, Hdim);

  // 1) QKV projection: [8,6144] = x @ Wqkv^T + b
  attn_gemm_xwT_partial<<<dim3(QKV_N / 128, KSPLIT), 256, 0, stream>>>(x_pad, wqkv, qkv_part, QKV_N);
  attn_reduce_bias<<<(Bsz * QKV_N + 255) / 256, 256, 0, stream>>>(qkv_part, bqkv, qkv_pad, QKV_N);

  // 2) Flash-decoding attention over the 8192-entry cache
  attn_split_kernel<<<dim3(NH, NSPLIT / 8), 256, 0, stream>>>(qkv_pad, ck, cv, wm, wl, wacc);
  attn_merge_kernel<<<NH * Bsz, HD, 0, stream>>>(qkv_pad, wm, wl, wacc, aout_pad);

  // 3) Output projection: [8,2048] = attn_out @ Wproj^T + b
  attn_gemm_xwT_partial<<<dim3(Hdim / 128, KSPLIT), 256, 0, stream>>>(aout_pad, wproj, proj_part, Hdim);
  attn_reduce_bias<<<(Bsz * Hdim + 255) / 256, 256, 0, stream>>>(proj_part, bproj, (float*)d_out, Hdim);
}